// GraphLearner_17025250362058
// MI455X (gfx1250) — compile-verified
//
#include <hip/hip_runtime.h>
#include <hip/hip_bf16.h>
#include <math.h>

// CDNA5 (gfx1250) wave32 WMMA types
typedef __attribute__((ext_vector_type(2))) float v2f;  // f32 A/B fragment (16x4 / 4x16)
typedef __attribute__((ext_vector_type(8))) float v8f;  // f32 C/D fragment (16x16)

#define N_NODES 8192
#define TOPK 32

#define BM 64    // workgroup M tile
#define BN 128   // workgroup N tile
#define BK 16    // K chunk
#define LDA 20   // padded LDS row stride in floats (80 B: 16B-aligned, conflict-free)

// Issue one wave-level async global->LDS B128 copy (16 B per lane, ASYNCcnt).
__device__ __forceinline__ void async_copy_b128(unsigned lds_off, const float* gptr) {
  asm volatile("global_load_async_to_lds_b128 %0, %1, off"
               :
               : "v"(lds_off), "v"(gptr)
               : "memory");
}

__device__ __forceinline__ void wait_async_all() {
  asm volatile("s_wait_asynccnt 0" ::: "memory");
}

// ---------------------------------------------------------------------------
// simgemm: S[r, c] = relu(E[rowBase+r, :] . E[c, :]) ; S[r,c] = -inf on diag.
// 256 threads = 8 wave32s in a 2(M) x 4(N) grid; each wave accumulates a
// 32x32 subtile as 2x2 of 16x16 via V_WMMA_F32_16X16X4_F32 (fp32-exact).
// Global->LDS staging uses async-to-LDS with double buffering.
// ---------------------------------------------------------------------------
__global__ __launch_bounds__(256) void simgemm_kernel(const float* __restrict__ E,
                                                      float* __restrict__ S,
                                                      int rowBase) {
  __shared__ float As[2][BM * LDA];  // 2 x 64 rows x 16 k (padded)
  __shared__ float Bs[2][BN * LDA];  // 2 x 128 cols x 16 k (padded)

  const int tid  = threadIdx.x;
  const int wave = tid >> 5;
  const int lane = tid & 31;
  const int half = lane >> 4;   // 0: lanes 0-15, 1: lanes 16-31
  const int l16  = lane & 15;

  const int mwave = (wave >> 2) * 32;  // 0,32
  const int nwave = (wave & 3) * 32;   // 0,32,64,96

  const int blockRow = blockIdx.x * BM;      // row offset within this chunk
  const int gRow0 = rowBase + blockRow;      // global first row of tile
  const int gCol0 = blockIdx.y * BN;         // global first col of tile

  // Cooperative loader mapping (one float4 lane-slot each):
  //  A: 64 rows x 4 quads = 256 slots -> 1 per thread
  //  B: 128 rows x 4 quads = 512 slots -> 2 per thread
  const int am = tid >> 2;          // 0..63
  const int aq = (tid & 3) * 4;     // k quad
  const int bm0 = tid >> 2;         // 0..63
  const int bm1 = bm0 + 64;         // 64..127

  const float* gA  = E + (size_t)(gRow0 + am)  * N_NODES + aq;
  const float* gB0 = E + (size_t)(gCol0 + bm0) * N_NODES + aq;
  const float* gB1 = E + (size_t)(gCol0 + bm1) * N_NODES + aq;

  // LDS byte offsets for this thread's staging slots (generic LDS pointer
  // truncates to the in-group offset on gfx1250's flat LDS aperture).
  unsigned ldsA[2], ldsB0[2], ldsB1[2];
#pragma unroll
  for (int b = 0; b < 2; ++b) {
    ldsA[b]  = (unsigned)(size_t)&As[b][am * LDA + aq];
    ldsB0[b] = (unsigned)(size_t)&Bs[b][bm0 * LDA + aq];
    ldsB1[b] = (unsigned)(size_t)&Bs[b][bm1 * LDA + aq];
  }

  v8f c00 = {}, c01 = {}, c10 = {}, c11 = {};

  // Prologue: stage chunk 0 into buffer 0.
  async_copy_b128(ldsA[0],  gA);
  async_copy_b128(ldsB0[0], gB0);
  async_copy_b128(ldsB1[0], gB1);
  wait_async_all();
  __syncthreads();

  int buf = 0;
  for (int k0 = 0; k0 < N_NODES; k0 += BK, buf ^= 1) {
    // Kick off next chunk into the other buffer while we compute this one.
    if (k0 + BK < N_NODES) {
      const int kn = k0 + BK;
      async_copy_b128(ldsA[buf ^ 1],  gA + kn);
      async_copy_b128(ldsB0[buf ^ 1], gB0 + kn);
      async_copy_b128(ldsB1[buf ^ 1], gB1 + kn);
    }

    const float* Ab = &As[buf][0];
    const float* Bb = &Bs[buf][0];
#pragma unroll
    for (int kk = 0; kk < BK; kk += 4) {
      // A 16x4 f32 layout: VGPR0 = K+2*half, VGPR1 = K+2*half+1, M = l16.
      const int ak = kk + 2 * half;
      const float2 a0f = *reinterpret_cast<const float2*>(&Ab[(mwave +  0 + l16) * LDA + ak]);
      const float2 a1f = *reinterpret_cast<const float2*>(&Ab[(mwave + 16 + l16) * LDA + ak]);
      const float2 b0f = *reinterpret_cast<const float2*>(&Bb[(nwave +  0 + l16) * LDA + ak]);
      const float2 b1f = *reinterpret_cast<const float2*>(&Bb[(nwave + 16 + l16) * LDA + ak]);
      v2f a0; a0.x = a0f.x; a0.y = a0f.y;
      v2f a1; a1.x = a1f.x; a1.y = a1f.y;
      v2f b0; b0.x = b0f.x; b0.y = b0f.y;
      v2f b1; b1.x = b1f.x; b1.y = b1f.y;
      c00 = __builtin_amdgcn_wmma_f32_16x16x4_f32(false, a0, false, b0, (short)0, c00, false, false);
      c01 = __builtin_amdgcn_wmma_f32_16x16x4_f32(false, a0, false, b1, (short)0, c01, false, false);
      c10 = __builtin_amdgcn_wmma_f32_16x16x4_f32(false, a1, false, b0, (short)0, c10, false, false);
      c11 = __builtin_amdgcn_wmma_f32_16x16x4_f32(false, a1, false, b1, (short)0, c11, false, false);
    }

    // Next chunk landed (this wave's asyncs) + everyone done reading `buf`.
    wait_async_all();
    __syncthreads();
  }

  // Epilogue. C/D layout: VGPR v -> row +v (lanes 0-15) / +v+8 (lanes 16-31),
  // col = subtile base + l16. ReLU then diagonal -inf, store to workspace.
  v8f* acc[2][2] = {{&c00, &c01}, {&c10, &c11}};
#pragma unroll
  for (int mi = 0; mi < 2; ++mi) {
#pragma unroll
    for (int ni = 0; ni < 2; ++ni) {
      const v8f cv = *acc[mi][ni];
      const int col = gCol0 + nwave + ni * 16 + l16;
#pragma unroll
      for (int v = 0; v < 8; ++v) {
        const int mloc = mwave + mi * 16 + v + 8 * half;
        const int grow = gRow0 + mloc;
        float val = fmaxf(cv[v], 0.0f);
        if (grow == col) val = -INFINITY;
        S[(size_t)(blockRow + mloc) * N_NODES + col] = val;
      }
    }
  }
}

// ---------------------------------------------------------------------------
// topk: one workgroup per row. Row cached in LDS (32 KB); 32 iterations of
// cooperative arg-max (tie-break: lowest index, matching jax.lax.top_k),
// winner overwritten with -inf each round.
// ---------------------------------------------------------------------------
__global__ __launch_bounds__(256) void topk_kernel(const float* __restrict__ S,
                                                   int rowBase,
                                                   int* __restrict__ ei_src,
                                                   int* __restrict__ ei_dst,
                                                   float* __restrict__ ew) {
  __shared__ float rowv[N_NODES];
  __shared__ float redV[256];
  __shared__ int   redI[256];

  const int tid = threadIdx.x;
  const int row = rowBase + blockIdx.x;
  const float* src = S + (size_t)blockIdx.x * N_NODES;

#pragma unroll
  for (int j = 0; j < N_NODES / 256; ++j)
    rowv[tid + j * 256] = src[tid + j * 256];
  __syncthreads();

  for (int sel = 0; sel < TOPK; ++sel) {
    float bv = -INFINITY;
    int bi = 0x7FFFFFFF;
#pragma unroll
    for (int j = 0; j < N_NODES / 256; ++j) {
      const int i = tid + j * 256;             // bank-conflict-free in wave32
      const float v = rowv[i];
      if (v > bv) { bv = v; bi = i; }          // ascending i => lowest-index tie-break
    }
    redV[tid] = bv;
    redI[tid] = bi;
    __syncthreads();
    for (int s = 128; s > 0; s >>= 1) {
      if (tid < s) {
        const float ov = redV[tid + s];
        const int   oi = redI[tid + s];
        if (ov > redV[tid] || (ov == redV[tid] && oi < redI[tid])) {
          redV[tid] = ov;
          redI[tid] = oi;
        }
      }
      __syncthreads();
    }
    if (tid == 0) {
      const int i = redI[0];
      ew[(size_t)row * TOPK + sel]     = redV[0];
      ei_dst[(size_t)row * TOPK + sel] = i;
      ei_src[(size_t)row * TOPK + sel] = row;
      rowv[i] = -INFINITY;
    }
    __syncthreads();
  }
}

// ---------------------------------------------------------------------------
// Launch: chunk rows by workspace size (one sim row = 32 KB); per chunk run
// fp32 WMMA GEMM into d_ws, then per-row top-k into d_out.
// d_out layout: int32 edge_index [2, N*K] flat, then float edge_weight [N*K].
// ---------------------------------------------------------------------------
extern "C" void kernel_launch(void* const* d_in, const int* in_sizes, int n_in,
                              void* d_out, int out_size, void* d_ws, size_t ws_size,
                              hipStream_t stream) {
  (void)in_sizes; (void)n_in; (void)out_size;
  const float* E = (const float*)d_in[0];

  const int NK = N_NODES * TOPK;
  int*   ei_src = (int*)d_out;           // edge_index[0, :]
  int*   ei_dst = (int*)d_out + NK;      // edge_index[1, :]
  float* ew     = (float*)d_out + 2 * NK;
  float* S      = (float*)d_ws;

  const size_t rowBytes = (size_t)N_NODES * sizeof(float);
  long maxRows = (long)(ws_size / rowBytes);
  if (maxRows > N_NODES) maxRows = N_NODES;
  maxRows = (maxRows / BM) * BM;       // gemm tile is BM rows
  if (maxRows < BM) maxRows = BM;      // needs >= 2 MB workspace

  for (int r0 = 0; r0 < N_NODES; r0 += (int)maxRows) {
    int rows = N_NODES - r0;
    if (rows > maxRows) rows = (int)maxRows;
    dim3 ggrid(rows / BM, N_NODES / BN);
    simgemm_kernel<<<ggrid, 256, 0, stream>>>(E, S, r0);
    topk_kernel<<<rows, 256, 0, stream>>>(S, r0, ei_src, ei_dst, ew);
  }
}